// Net_42880953483581
// MI455X (gfx1250) — compile-verified
//
#include <hip/hip_runtime.h>

#define N_NODES 50000
#define N_EDGES 600000
#define DIM     128
#define N_REL   8
#define A_STRIDE 132   // LDS row stride in floats: 132 % 64 = 4 -> conflict-free b64 reads

typedef __attribute__((ext_vector_type(2))) float v2f;
typedef __attribute__((ext_vector_type(8))) float v8f;

// ---------------------------------------------------------------------------
// Kernel 1: zero workspace (sums [R*N*D] + cnts [R*N]) as float4s
// ---------------------------------------------------------------------------
__global__ void zero_ws_kernel(float4* __restrict__ ws, long n4) {
  long i = (long)blockIdx.x * blockDim.x + threadIdx.x;
  long stride = (long)gridDim.x * blockDim.x;
  float4 z = {0.f, 0.f, 0.f, 0.f};
  for (; i < n4; i += stride) ws[i] = z;
}

// ---------------------------------------------------------------------------
// Kernel 2: per-edge scatter-add.  One wave32 per edge: each lane moves a
// float4 of x[src] (L2-resident: x is 25.6 MB < 192 MB L2) into
// sums[(r*N+dst)*D] via hardware global_atomic_add_f32.
// ---------------------------------------------------------------------------
__global__ void __launch_bounds__(256)
scatter_kernel(const float* __restrict__ x,
               const int*   __restrict__ edge_index,
               const int*   __restrict__ edge_type,
               float* __restrict__ sums,
               float* __restrict__ cnts) {
  const int wave = threadIdx.x >> 5;
  const int lane = threadIdx.x & 31;
  const long e = (long)blockIdx.x * 8 + wave;
  if (e >= N_EDGES) return;

  const int src = edge_index[e];
  const int dst = edge_index[N_EDGES + e];
  const int rt  = edge_type[e];

  const float4 v = ((const float4*)(x + (size_t)src * DIM))[lane];
  float* s = sums + ((size_t)rt * N_NODES + (size_t)dst) * DIM + lane * 4;
  unsafeAtomicAdd(s + 0, v.x);
  unsafeAtomicAdd(s + 1, v.y);
  unsafeAtomicAdd(s + 2, v.z);
  unsafeAtomicAdd(s + 3, v.w);
  if (lane == 0) unsafeAtomicAdd(cnts + (size_t)rt * N_NODES + dst, 1.0f);
}

// ---------------------------------------------------------------------------
// Kernel 3: fused  out = relu( sum_r mean_r @ W_r + x @ root + bias )
// Block = 16 nodes x 128 outputs; 8 waves, each owns a 16x16 output tile and
// accumulates K = 9*128 with V_WMMA_F32_16X16X4_F32 (exact fp32 path).
// ---------------------------------------------------------------------------
__global__ void __launch_bounds__(256)
rgcn_wmma_kernel(const float* __restrict__ sums,
                 const float* __restrict__ cnts,
                 const float* __restrict__ x,
                 const float* __restrict__ weight,
                 const float* __restrict__ root,
                 const float* __restrict__ bias,
                 float* __restrict__ out) {
  __shared__ float sA[16 * A_STRIDE];

  const int nodeBase = blockIdx.x * 16;
  const int tid  = threadIdx.x;
  const int lane = tid & 31;
  const int wv   = tid >> 5;       // wave id 0..7 -> output column tile
  const int col0 = wv * 16;
  const int half = lane >> 4;      // lane group (0: lanes 0-15, 1: lanes 16-31)
  const int m    = lane & 15;      // row (A) / column (B,C) within tile

  // staging coordinates: thread t loads 8 consecutive floats of one A row
  const int si = tid >> 4;         // A row 0..15
  const int sc = (tid & 15) * 8;   // A col start

  v8f acc = {};

  for (int rr = 0; rr < 9; ++rr) {
    __syncthreads();               // previous tile fully consumed

    // ---- stage 16x128 A tile (scaled relation mean, or x for root term) ----
    float vals[8];
    if (rr < 8) {
      const float cnt = cnts[(size_t)rr * N_NODES + nodeBase + si];
      const float inv = 1.0f / fmaxf(cnt, 1.0f);
      const float* srow =
          sums + ((size_t)rr * N_NODES + (size_t)(nodeBase + si)) * DIM + sc;
      const float4 a0 = *(const float4*)(srow);
      const float4 a1 = *(const float4*)(srow + 4);
      vals[0] = a0.x * inv; vals[1] = a0.y * inv;
      vals[2] = a0.z * inv; vals[3] = a0.w * inv;
      vals[4] = a1.x * inv; vals[5] = a1.y * inv;
      vals[6] = a1.z * inv; vals[7] = a1.w * inv;
    } else {
      const float* srow = x + (size_t)(nodeBase + si) * DIM + sc;
      const float4 a0 = *(const float4*)(srow);
      const float4 a1 = *(const float4*)(srow + 4);
      vals[0] = a0.x; vals[1] = a0.y; vals[2] = a0.z; vals[3] = a0.w;
      vals[4] = a1.x; vals[5] = a1.y; vals[6] = a1.z; vals[7] = a1.w;
    }
#pragma unroll
    for (int j = 0; j < 8; ++j) sA[si * A_STRIDE + sc + j] = vals[j];
    __syncthreads();

    // ---- 32 x WMMA f32 16x16x4 over K = 0..127 ----
    const float* __restrict__ B =
        (rr < 8) ? (weight + (size_t)rr * DIM * DIM) : root;
    const float* sArow = sA + m * A_STRIDE;

#pragma unroll 4
    for (int k = 0; k < DIM; k += 4) {
      const int ka = k + 2 * half;   // ISA 16x4 f32 layout: VGPR0 = K0|K2, VGPR1 = K1|K3
      v2f a, b;
      a.x = sArow[ka];
      a.y = sArow[ka + 1];
      b.x = B[(size_t)ka * DIM + col0 + m];
      b.y = B[(size_t)(ka + 1) * DIM + col0 + m];
      acc = __builtin_amdgcn_wmma_f32_16x16x4_f32(
          /*neg_a=*/false, a, /*neg_b=*/false, b,
          /*c_mod=*/(short)0, acc, /*reuse_a=*/false, /*reuse_b=*/false);
    }
  }

  // ---- epilogue: bias + ReLU, C layout VGPR j -> rows j / j+8 ----
  const float bcol = bias[col0 + m];
#pragma unroll
  for (int j = 0; j < 8; ++j) {
    const int row = j + 8 * half;
    float v = acc[j] + bcol;
    v = v > 0.0f ? v : 0.0f;
    out[(size_t)(nodeBase + row) * DIM + col0 + m] = v;
  }
}

// ---------------------------------------------------------------------------
extern "C" void kernel_launch(void* const* d_in, const int* in_sizes, int n_in,
                              void* d_out, int out_size, void* d_ws, size_t ws_size,
                              hipStream_t stream) {
  const float* x          = (const float*)d_in[0];
  const int*   edge_index = (const int*)  d_in[1];
  const int*   edge_type  = (const int*)  d_in[2];
  const float* weight     = (const float*)d_in[3];
  const float* root       = (const float*)d_in[4];
  const float* bias       = (const float*)d_in[5];
  float* out = (float*)d_out;

  float* sums = (float*)d_ws;                                  // R*N*D floats
  float* cnts = sums + (size_t)N_REL * N_NODES * DIM;          // R*N floats

  const long n4 = ((long)N_REL * N_NODES * DIM + (long)N_REL * N_NODES) / 4;
  zero_ws_kernel<<<4096, 256, 0, stream>>>((float4*)d_ws, n4);

  scatter_kernel<<<(N_EDGES + 7) / 8, 256, 0, stream>>>(
      x, edge_index, edge_type, sums, cnts);

  rgcn_wmma_kernel<<<N_NODES / 16, 256, 0, stream>>>(
      sums, cnts, x, weight, root, bias, out);
}